// Net_38766374813749
// MI455X (gfx1250) — compile-verified
//
#include <hip/hip_runtime.h>
#include <hip/hip_bf16.h>

#define NNODES 50000
#define NEDGES 1600000
#define NFEAT  512
#define F1     128   // heads1 * C = 8*16
#define H1_    8
#define C_     16    // out channels per head (both layers)
#define NCLASS 16

typedef float v2f __attribute__((ext_vector_type(2)));
typedef float v8f __attribute__((ext_vector_type(8)));

// Order-preserving float -> uint encoding so atomicMax(uint) == float max.
__device__ __forceinline__ unsigned enc_f32(float f) {
  unsigned u = __float_as_uint(f);
  return (u & 0x80000000u) ? ~u : (u | 0x80000000u);
}
__device__ __forceinline__ float dec_f32(unsigned u) {
  return (u & 0x80000000u) ? __uint_as_float(u & 0x7FFFFFFFu) : __uint_as_float(~u);
}

// ---------------------------------------------------------------------------
// D[M,NOUT] = X[M,K] @ Wt[NOUT,K]^T  using V_WMMA_F32_16X16X4_F32.
// One wave computes a 16-row x (TN*16)-col strip: the A fragment is loaded
// once per k-step and reused for TN WMMAs (TN accumulators in registers).
// K/NOUT are template constants so every load is base + immediate offset
// (clause-able, no address chains). Loads are hoisted ahead of the WMMA
// block so s_wait_loadcnt can stagger. M % 16 == 0 -> EXEC all 1s.
// A frag (16x4 f32): lanes 0-15 hold K = k,k+1 ; lanes 16-31 hold K = k+2,k+3.
// B frag (4x16 f32): symmetric layout, n = lane&15.
// C/D: VGPR r -> row (half*8 + r), col = lane&15.
// ---------------------------------------------------------------------------
template <int TN, int K, int NOUT>
__global__ __launch_bounds__(32) void gemm_f32_wmma_strip(
    const float* __restrict__ X, const float* __restrict__ Wt,
    float* __restrict__ D) {
  int tm = blockIdx.x;
  int lane = threadIdx.x;
  int l15 = lane & 15, half = lane >> 4;
  const float* ap = X  + (size_t)(tm * 16 + l15) * K + half * 2;
  const float* bp = Wt + (size_t)l15 * K + half * 2;
  v8f acc[TN];
  #pragma unroll
  for (int t = 0; t < TN; ++t) acc[t] = (v8f){};
  #pragma unroll 2
  for (int k = 0; k < K; k += 4) {
    // ---- load phase: 1 A fragment + TN B fragments, immediate offsets ----
    v2f a, b[TN];
    a.x = ap[k];
    a.y = ap[k + 1];
    #pragma unroll
    for (int t = 0; t < TN; ++t) {
      b[t].x = bp[t * 16 * K + k];
      b[t].y = bp[t * 16 * K + k + 1];
    }
    // ---- compute phase: TN independent accumulators ----
    #pragma unroll
    for (int t = 0; t < TN; ++t)
      acc[t] = __builtin_amdgcn_wmma_f32_16x16x4_f32(false, a, false, b[t],
                                                     (short)0, acc[t], false, false);
  }
  #pragma unroll
  for (int t = 0; t < TN; ++t) {
    float* dp = D + (size_t)(tm * 16 + half * 8) * NOUT + t * 16 + l15;
    #pragma unroll
    for (int r = 0; r < 8; ++r) dp[r * NOUT] = acc[t][r];
  }
}

// a_src[n,h] = sum_c h[n,h,c]*attS[h,c] ; a_dst likewise. One thread per (n,h).
__global__ void attn_scores(const float* __restrict__ h,
                            const float* __restrict__ attS,
                            const float* __restrict__ attD,
                            float* __restrict__ aS, float* __restrict__ aD,
                            int NH, int H) {
  int t = blockIdx.x * blockDim.x + threadIdx.x;
  if (t >= NH) return;
  int hh = t % H;
  const float* hp = h + (size_t)t * C_;
  const float* sp = attS + hh * C_;
  const float* dp = attD + hh * C_;
  float s = 0.f, d = 0.f;
  #pragma unroll
  for (int c = 0; c < C_; ++c) { s += hp[c] * sp[c]; d += hp[c] * dp[c]; }
  aS[t] = s; aD[t] = d;
}

__global__ void fill_u32(unsigned* __restrict__ p, unsigned v, size_t n) {
  size_t t = (size_t)blockIdx.x * blockDim.x + threadIdx.x;
  if (t < n) p[t] = v;
}

// Pass 1: segment max of leaky_relu(a_src[src]+a_dst[dst]) into amax[dst,h].
template <int H>
__global__ void edge_max(const int* __restrict__ ei,
                         const float* __restrict__ aS, const float* __restrict__ aD,
                         unsigned* __restrict__ amax) {
  int e = blockIdx.x * blockDim.x + threadIdx.x;
  if (e >= NEDGES + NNODES) return;
  int s, d;
  if (e < NEDGES) { s = ei[e]; d = ei[NEDGES + e]; } else { s = d = e - NEDGES; }
  const float* sp = aS + (size_t)s * H;
  const float* dp = aD + (size_t)d * H;
  unsigned* mp = amax + (size_t)d * H;
  #pragma unroll
  for (int h = 0; h < H; ++h) {
    float al = sp[h] + dp[h];
    al = al > 0.f ? al : 0.2f * al;
    atomicMax(&mp[h], enc_f32(al));
  }
}

// Pass 2: denom[dst,h] += exp(alpha - amax[dst,h])
template <int H>
__global__ void edge_sum(const int* __restrict__ ei,
                         const float* __restrict__ aS, const float* __restrict__ aD,
                         const unsigned* __restrict__ amax, float* __restrict__ denom) {
  int e = blockIdx.x * blockDim.x + threadIdx.x;
  if (e >= NEDGES + NNODES) return;
  int s, d;
  if (e < NEDGES) { s = ei[e]; d = ei[NEDGES + e]; } else { s = d = e - NEDGES; }
  const float* sp = aS + (size_t)s * H;
  const float* dp = aD + (size_t)d * H;
  const unsigned* mp = amax + (size_t)d * H;
  float* np = denom + (size_t)d * H;
  #pragma unroll
  for (int h = 0; h < H; ++h) {
    float al = sp[h] + dp[h];
    al = al > 0.f ? al : 0.2f * al;
    atomicAdd(&np[h], expf(al - dec_f32(mp[h])));
  }
}

// Pass 3: agg[dst,h,:] += coeff * h[src,h,:]  ; one thread per (edge, head).
template <int H>
__global__ void edge_msg(const int* __restrict__ ei,
                         const float* __restrict__ aS, const float* __restrict__ aD,
                         const unsigned* __restrict__ amax, const float* __restrict__ denom,
                         const float* __restrict__ h, float* __restrict__ agg) {
  long long t = (long long)blockIdx.x * blockDim.x + threadIdx.x;
  long long total = (long long)(NEDGES + NNODES) * H;
  if (t >= total) return;
  int hh = (int)(t % H);
  int e  = (int)(t / H);
  int s, d;
  if (e < NEDGES) { s = ei[e]; d = ei[NEDGES + e]; } else { s = d = e - NEDGES; }
  size_t si = (size_t)s * H + hh, di = (size_t)d * H + hh;
  float al = aS[si] + aD[di];
  al = al > 0.f ? al : 0.2f * al;
  float coeff = expf(al - dec_f32(amax[di])) / (denom[di] + 1e-16f);
  const float* hp = h + si * C_;
  float* op = agg + di * C_;
  #pragma unroll
  for (int c = 0; c < C_; ++c) atomicAdd(&op[c], hp[c] * coeff);
}

// x[t] = elu(x[t] + b[t % F])  (in place)
__global__ void bias_elu(float* __restrict__ x, const float* __restrict__ b,
                         int F, size_t n) {
  size_t t = (size_t)blockIdx.x * blockDim.x + threadIdx.x;
  if (t >= n) return;
  float v = x[t] + b[(int)(t % F)];
  x[t] = v > 0.f ? v : expm1f(v);
}

// out[n,:] = log_softmax(agg[n,:] + b)
__global__ void bias_logsoftmax(const float* __restrict__ agg,
                                const float* __restrict__ b,
                                float* __restrict__ out) {
  int n = blockIdx.x * blockDim.x + threadIdx.x;
  if (n >= NNODES) return;
  float v[NCLASS];
  float m = -1e30f;
  #pragma unroll
  for (int c = 0; c < NCLASS; ++c) {
    v[c] = agg[(size_t)n * NCLASS + c] + b[c];
    m = fmaxf(m, v[c]);
  }
  float s = 0.f;
  #pragma unroll
  for (int c = 0; c < NCLASS; ++c) s += expf(v[c] - m);
  float ls = logf(s);
  #pragma unroll
  for (int c = 0; c < NCLASS; ++c) out[(size_t)n * NCLASS + c] = v[c] - m - ls;
}

extern "C" void kernel_launch(void* const* d_in, const int* in_sizes, int n_in,
                              void* d_out, int out_size, void* d_ws, size_t ws_size,
                              hipStream_t stream) {
  const float* x     = (const float*)d_in[0];
  const int*   ei    = (const int*)  d_in[1];
  const float* W1    = (const float*)d_in[2];
  const float* attS1 = (const float*)d_in[3];
  const float* attD1 = (const float*)d_in[4];
  const float* b1    = (const float*)d_in[5];
  const float* W2    = (const float*)d_in[6];
  const float* attS2 = (const float*)d_in[7];
  const float* attD2 = (const float*)d_in[8];
  const float* b2    = (const float*)d_in[9];
  float* out = (float*)d_out;

  char* ws = (char*)d_ws;
  size_t off = 0;
  auto alloc = [&](size_t bytes) -> void* {
    void* p = ws + off;
    off = (off + bytes + 255) & ~(size_t)255;
    return p;
  };

  float*    h1   = (float*)   alloc((size_t)NNODES * F1 * 4);
  float*    as1  = (float*)   alloc((size_t)NNODES * H1_ * 4);
  float*    ad1  = (float*)   alloc((size_t)NNODES * H1_ * 4);
  unsigned* am1  = (unsigned*)alloc((size_t)NNODES * H1_ * 4);
  float*    dn1  = (float*)   alloc((size_t)NNODES * H1_ * 4);
  float*    agg1 = (float*)   alloc((size_t)NNODES * F1 * 4);   // reused as ELU output
  float*    h2   = (float*)   alloc((size_t)NNODES * C_ * 4);
  float*    as2  = (float*)   alloc((size_t)NNODES * 4);
  float*    ad2  = (float*)   alloc((size_t)NNODES * 4);
  unsigned* am2  = (unsigned*)alloc((size_t)NNODES * 4);
  float*    dn2  = (float*)   alloc((size_t)NNODES * 4);
  float*    agg2 = (float*)   alloc((size_t)NNODES * C_ * 4);

  const int EDG = NEDGES + NNODES;
  const int TB = 256;
  auto nb = [](long long n, int b) { return (unsigned)((n + b - 1) / b); };

  // ---- Layer 1 ----
  // h1 = x @ W1^T   (M=50000, K=512, Nout=128): 3125 waves, each a 16x128 strip.
  gemm_f32_wmma_strip<8, NFEAT, F1><<<3125, 32, 0, stream>>>(x, W1, h1);
  attn_scores<<<nb((long long)NNODES * H1_, TB), TB, 0, stream>>>(
      h1, attS1, attD1, as1, ad1, NNODES * H1_, H1_);
  fill_u32<<<nb((long long)NNODES * H1_, TB), TB, 0, stream>>>(am1, 0u, (size_t)NNODES * H1_);
  fill_u32<<<nb((long long)NNODES * H1_, TB), TB, 0, stream>>>((unsigned*)dn1, 0u, (size_t)NNODES * H1_);
  fill_u32<<<nb((long long)NNODES * F1, TB), TB, 0, stream>>>((unsigned*)agg1, 0u, (size_t)NNODES * F1);
  edge_max<H1_><<<nb(EDG, TB), TB, 0, stream>>>(ei, as1, ad1, am1);
  edge_sum<H1_><<<nb(EDG, TB), TB, 0, stream>>>(ei, as1, ad1, am1, dn1);
  edge_msg<H1_><<<nb((long long)EDG * H1_, TB), TB, 0, stream>>>(
      ei, as1, ad1, am1, dn1, h1, agg1);
  bias_elu<<<nb((long long)NNODES * F1, TB), TB, 0, stream>>>(
      agg1, b1, F1, (size_t)NNODES * F1);

  // ---- Layer 2 ----
  // h2 = elu_out @ W2^T   (M=50000, K=128, Nout=16)
  gemm_f32_wmma_strip<1, F1, C_><<<3125, 32, 0, stream>>>(agg1, W2, h2);
  attn_scores<<<nb(NNODES, TB), TB, 0, stream>>>(h2, attS2, attD2, as2, ad2, NNODES, 1);
  fill_u32<<<nb(NNODES, TB), TB, 0, stream>>>(am2, 0u, (size_t)NNODES);
  fill_u32<<<nb(NNODES, TB), TB, 0, stream>>>((unsigned*)dn2, 0u, (size_t)NNODES);
  fill_u32<<<nb((long long)NNODES * C_, TB), TB, 0, stream>>>((unsigned*)agg2, 0u, (size_t)NNODES * C_);
  edge_max<1><<<nb(EDG, TB), TB, 0, stream>>>(ei, as2, ad2, am2);
  edge_sum<1><<<nb(EDG, TB), TB, 0, stream>>>(ei, as2, ad2, am2, dn2);
  edge_msg<1><<<nb(EDG, TB), TB, 0, stream>>>(ei, as2, ad2, am2, dn2, h2, agg2);

  bias_logsoftmax<<<nb(NNODES, TB), TB, 0, stream>>>(agg2, b2, out);
}